// QuadtreeEmbedding_64166811402729
// MI455X (gfx1250) — compile-verified
//
#include <hip/hip_runtime.h>
#include <hip/hip_bf16.h>

#define PATCHES_TOTAL 1365
#define K_DIM 12288
#define E_DIM 512
#define EPSF 1e-5f

typedef float v2f __attribute__((ext_vector_type(2)));
typedef float v8f __attribute__((ext_vector_type(8)));
typedef unsigned int u32x4 __attribute__((ext_vector_type(4)));
typedef int i32x8 __attribute__((ext_vector_type(8)));
typedef int i32x4 __attribute__((ext_vector_type(4)));

// ---------------------------------------------------------------------------
// Tensor Data Mover descriptor build + issue (CDNA5 ISA).
// Up-to-3D tile load, data_size = 4 B. pad32: +1 dword LDS pad per 32 dwords
// (interval code 4 = 32 dwords, amount code 0 = 1 dword) -> stride-33 rows.
// OOB reads beyond tensor_dim return zero (free edge guard).
// lds_off: low 32 bits of a generic pointer to LDS == wave-relative LDS byte
// offset (aperture bits live in [63:32]).
// clang-23 toolchain: 6-arg builtin (extra i32x8 group, zero-filled).
// ---------------------------------------------------------------------------
static __device__ inline void qt_tdm_load(
    unsigned lds_off, const void* gptr,
    unsigned t_dim0, unsigned t_dim1, unsigned t_dim2,
    unsigned tile0, unsigned tile1, unsigned tile2,
    unsigned long long stride0, unsigned long long stride1, bool pad32)
{
    unsigned long long ga = (unsigned long long)(size_t)gptr;
    u32x4 g0;
    g0.x = 1u;                                    // count=1 (valid user D#)
    g0.y = lds_off;                               // lds_addr (bytes)
    g0.z = (unsigned)(ga & 0xFFFFFFFFu);          // global_addr[31:0]
    g0.w = (unsigned)((ga >> 32) & 0x01FFFFFFu)   // global_addr[56:32]
         | (2u << 30);                            // type = 2 ("image")

    i32x8 g1;
    unsigned w0 = (2u << 16);                     // data_size = 4 B
    if (pad32) w0 |= (1u << 20) | (4u << 22);     // pad_enable, 32dw/1dw
    g1[0] = (int)w0;                              // wg_mask=0, flags
    g1[1] = (int)((t_dim0 & 0xFFFFu) << 16);      // [63:48]=dim0 lo16
    g1[2] = (int)(((t_dim0 >> 16) & 0xFFFFu) | ((t_dim1 & 0xFFFFu) << 16));
    g1[3] = (int)(((t_dim1 >> 16) & 0xFFFFu) | ((tile0 & 0xFFFFu) << 16));
    g1[4] = (int)((tile1 & 0xFFFFu) | ((tile2 & 0xFFFFu) << 16));
    g1[5] = (int)(unsigned)(stride0 & 0xFFFFFFFFull);          // dim0_stride lo32
    g1[6] = (int)((unsigned)((stride0 >> 32) & 0xFFFFull)      // dim0_stride hi16
                | ((unsigned)(stride1 & 0xFFFFull) << 16));    // dim1_stride lo16
    g1[7] = (int)(unsigned)((stride1 >> 16) & 0xFFFFFFFFull);  // dim1_stride hi32

    i32x4 g2;
    g2[0] = (int)t_dim2;                          // tensor_dim2 (0 for 2D)
    g2[1] = 0; g2[2] = 0; g2[3] = 0;
    i32x4 g3 = {0, 0, 0, 0};
    i32x8 g4 = {0, 0, 0, 0, 0, 0, 0, 0};          // extra group (clang-23 form)
    __builtin_amdgcn_tensor_load_to_lds(g0, g1, g2, g3, g4, 0);
}

static __device__ inline unsigned qt_lds_off(const void* p)
{
    return (unsigned)(size_t)p;
}

// ---------------------------------------------------------------------------
// Kernel 1: extract all 1365 multi-resolution patches.
// nearest resize to n*64 from 2048 is exact stride sampling: src = dst << (5-l)
// ---------------------------------------------------------------------------
__global__ __launch_bounds__(256) void qt_extract_patches(
    const float* __restrict__ x, float* __restrict__ patches)
{
    int gid = blockIdx.x * 256 + threadIdx.x;
    if (gid >= PATCHES_TOTAL * K_DIM) return;
    int p  = gid / K_DIM;
    int r  = gid - p * K_DIM;
    int c  = r >> 12;
    int py = (r >> 6) & 63;
    int px = r & 63;

    int l = 0, base = 0, n = 1;
    while (l < 5) {
        int cnt = n * n;
        if (p < base + cnt) break;
        base += cnt; n <<= 1; ++l;
    }
    int local = p - base;
    int i = local / n, j = local - i * n;
    int sh = 5 - l;
    int sy = ((i << 6) + py) << sh;
    int sx = ((j << 6) + px) << sh;
    patches[gid] = x[((size_t)c << 22) + ((size_t)sy << 11) + (size_t)sx];
}

// ---------------------------------------------------------------------------
// Kernel 2: fused splitter CNN, one workgroup (256 threads = 8 waves) / patch.
// conv1/conv2 VALU; conv3 (72% of all FLOPs) as fp32 WMMA implicit GEMM
// (M=116 band pixels, N=64 ch, K=288 = tap*32+ci).  B weights pre-swizzled
// into pair-interleaved LDS w3p[k/2][n][2] so every frag is one ds_load_b64.
// Input band (3x8x64) streamed by the Tensor Data Mover, double-buffered:
// TDM copies band b+1 while conv pipeline runs band b.
// fp32 throughout: the logit SIGN gates the whole tree.
// ---------------------------------------------------------------------------
__global__ __launch_bounds__(256) void qt_splitter(
    const float* __restrict__ patches,
    const float* __restrict__ c1w, const float* __restrict__ c1b,
    const float* __restrict__ g1,  const float* __restrict__ b1,
    const float* __restrict__ m1,  const float* __restrict__ v1,
    const float* __restrict__ c2w, const float* __restrict__ c2b,
    const float* __restrict__ g2,  const float* __restrict__ b2,
    const float* __restrict__ m2,  const float* __restrict__ v2,
    const float* __restrict__ c3w, const float* __restrict__ c3b,
    const float* __restrict__ g3,  const float* __restrict__ b3,
    const float* __restrict__ m3,  const float* __restrict__ v3,
    const float* __restrict__ lw,  const float* __restrict__ lb,
    float* __restrict__ logits)
{
    extern __shared__ float smem[];
    float* w3p   = smem;                 // [144 kpair][64 n][2] = 18432 f (72 KB)
    float* s_in2 = w3p + 288 * 64;       // 2 x [3][8][64] = 3072 f
    float* s_h1  = s_in2 + 3072;         // [16][6][62]    = 5952 f
    float* s_h2  = s_h1 + 5952;          // [32][4][60]    = 7680 f
    float* s_red = s_h2 + 7680;          // [16][64]       = 1024 f
    float* s_fin = s_red + 1024;         // [64]

    const int t    = threadIdx.x;
    const int p    = blockIdx.x;
    const int lane = t & 31;
    const int wave = t >> 5;          // = pixel tile (0..7)
    const int lm   = lane & 15;
    const int hi   = lane >> 4;
    const float* patch = patches + (size_t)p * K_DIM;

    // stage conv3 weights, pair-interleaved: k = tap*32 + ci (tap = ky*3+kx),
    // w3p[(k>>1)*128 + n*2 + (k&1)] = c3w[n][ci][ky][kx]
    for (int idx = t; idx < 288 * 64; idx += 256) {
        int kpair = idx >> 7;
        int rem   = idx & 127;
        int n = rem >> 1, e = rem & 1;
        int k = kpair * 2 + e;
        int ci = k & 31, tap = k >> 5;
        w3p[idx] = c3w[n * 288 + ci * 9 + tap];
    }

    // preload band 0 input via TDM (3D tile: x=64 cols, y=8 rows, z=3 ch)
    if (wave == 0) {
        qt_tdm_load(qt_lds_off(s_in2), patch,
                    /*t_dim*/64, 64, 3, /*tile*/64, 8, 3,
                    /*strides*/64ull, 4096ull, false);
        __builtin_amdgcn_s_wait_tensorcnt(0);
    }
    __syncthreads();   // covers w3p staging + band-0 input

    // this lane's A-matrix pixel (clamped; rows >=116 are discarded later)
    const int apix = min(wave * 16 + lm, 115);
    const int arr  = apix / 58;
    const int axx  = apix - arr * 58;

    const v2f* w3pv = (const v2f*)w3p;   // pair view (8B aligned)
    float acc[4] = {0.f, 0.f, 0.f, 0.f};

    for (int band = 0; band < 29; ++band) {
        const float* sin = s_in2 + (band & 1) * 1536;

        // conv1 + relu + BN : 16 x 6 x 62 (VALU)
        for (int idx = t; idx < 16 * 6 * 62; idx += 256) {
            int co = idx / 372; int rem = idx - co * 372;
            int q = rem / 62;   int xx = rem - q * 62;
            float s = c1b[co];
            #pragma unroll
            for (int ci = 0; ci < 3; ++ci)
                #pragma unroll
                for (int ky = 0; ky < 3; ++ky)
                    #pragma unroll
                    for (int kx = 0; kx < 3; ++kx)
                        s += sin[ci * 512 + (q + ky) * 64 + xx + kx] *
                             c1w[((co * 3 + ci) * 3 + ky) * 3 + kx];
            s = fmaxf(s, 0.f);
            float inv = g1[co] * rsqrtf(v1[co] + EPSF);
            s_h1[co * 372 + q * 62 + xx] = s * inv + (b1[co] - m1[co] * inv);
        }
        __syncthreads();

        // kick off next band's input copy while we compute this band
        if (wave == 0 && band < 28) {
            int rn = (band + 1) * 2;
            qt_tdm_load(qt_lds_off(s_in2 + ((band + 1) & 1) * 1536),
                        patch + rn * 64,
                        64, (unsigned)(64 - rn), 3, 64, 8, 3,
                        64ull, 4096ull, false);
        }

        // conv2 + relu + BN : 32 x 4 x 60 (VALU)
        for (int idx = t; idx < 32 * 4 * 60; idx += 256) {
            int co = idx / 240; int rem = idx - co * 240;
            int q = rem / 60;   int xx = rem - q * 60;
            float s = c2b[co];
            for (int ci = 0; ci < 16; ++ci) {
                const float* w = c2w + ((co * 16 + ci) * 9);
                const float* h = s_h1 + ci * 372 + q * 62 + xx;
                s += h[0]   * w[0] + h[1]   * w[1] + h[2]   * w[2]
                   + h[62]  * w[3] + h[63]  * w[4] + h[64]  * w[5]
                   + h[124] * w[6] + h[125] * w[7] + h[126] * w[8];
            }
            s = fmaxf(s, 0.f);
            float inv = g2[co] * rsqrtf(v2[co] + EPSF);
            s_h2[co * 240 + q * 60 + xx] = s * inv + (b2[co] - m2[co] * inv);
        }
        __syncthreads();

        // conv3 via fp32 WMMA implicit GEMM: 288 WMMAs / wave / band
        v8f d[4];
        #pragma unroll
        for (int c = 0; c < 4; ++c) d[c] = (v8f){0,0,0,0,0,0,0,0};

        for (int tap = 0; tap < 9; ++tap) {
            const int ky = tap / 3;
            const int kx = tap - 3 * ky;
            const int rowoff = (arr + ky) * 60 + (axx + kx);
            #pragma unroll
            for (int cib = 0; cib < 32; cib += 4) {
                const int ciA = cib + 2 * hi;
                v2f a;
                a.x = s_h2[ciA * 240 + rowoff];
                a.y = s_h2[(ciA + 1) * 240 + rowoff];
                // pair index: (k>>1)*64 + n, k = tap*32 + ciA (even)
                const int kp = (tap * 16 + (ciA >> 1)) * 64 + lm;
                v2f fb0 = w3pv[kp];
                v2f fb1 = w3pv[kp + 16];
                v2f fb2 = w3pv[kp + 32];
                v2f fb3 = w3pv[kp + 48];
                d[0] = __builtin_amdgcn_wmma_f32_16x16x4_f32(
                           false, a, false, fb0, (short)0, d[0], false, false);
                d[1] = __builtin_amdgcn_wmma_f32_16x16x4_f32(
                           false, a, false, fb1, (short)0, d[1], false, false);
                d[2] = __builtin_amdgcn_wmma_f32_16x16x4_f32(
                           false, a, false, fb2, (short)0, d[2], false, false);
                d[3] = __builtin_amdgcn_wmma_f32_16x16x4_f32(
                           false, a, false, fb3, (short)0, d[3], false, false);
            }
        }

        // relu(conv3 + bias) -> per-channel partial sums
        #pragma unroll
        for (int c = 0; c < 4; ++c) {
            const float bias = c3b[c * 16 + lm];
            #pragma unroll
            for (int r = 0; r < 8; ++r) {
                int pixel = wave * 16 + r + 8 * hi;
                float v = d[c][r] + bias;
                acc[c] += (pixel < 116) ? fmaxf(v, 0.f) : 0.f;
            }
        }
        __syncthreads();                  // everyone done with s_h1/s_h2
        if (wave == 0) __builtin_amdgcn_s_wait_tensorcnt(0);
        __syncthreads();                  // next input buffer visible
    }

    // cross-wave reduction: 16 partials per channel
    {
        const int g = wave * 2 + hi;
        #pragma unroll
        for (int c = 0; c < 4; ++c)
            s_red[g * 64 + c * 16 + lm] = acc[c];
    }
    __syncthreads();
    if (t < 64) {
        float tot = 0.f;
        #pragma unroll
        for (int g = 0; g < 16; ++g) tot += s_red[g * 64 + t];
        float mean = tot * (1.f / (58.f * 58.f));
        float inv  = g3[t] * rsqrtf(v3[t] + EPSF);
        float h3   = mean * inv + (b3[t] - m3[t] * inv);
        s_fin[t]   = h3 * lw[t];
    }
    __syncthreads();
    if (t == 0) {
        float s = lb[0];
        for (int i = 0; i < 64; ++i) s += s_fin[i];
        logits[p] = s;
    }
}

// ---------------------------------------------------------------------------
// Kernel 3: embedding GEMM, emb[1365][512] = patches[1365][12288] * bw^T + bb
// fp32 WMMA; 4 waves/block, tile M16 x N64, KC=32.  A/B tiles streamed by the
// Tensor Data Mover (2D descriptors, row stride K, pad 1dw/32dw -> stride-33
// LDS rows), double-buffered: TDM fetches chunk k+1 during chunk k's WMMAs.
// tensor_dim1 = rows remaining, so TDM OOB zero-fill replaces the row guard.
// ---------------------------------------------------------------------------
__global__ __launch_bounds__(128) void qt_embed_gemm(
    const float* __restrict__ patches, const float* __restrict__ bw,
    const float* __restrict__ bb, float* __restrict__ emb)
{
    __shared__ float As[2][16 * 33];
    __shared__ float Bs[2][64 * 33];

    const int t    = threadIdx.x;
    const int lane = t & 31;
    const int wave = t >> 5;
    const int lm   = lane & 15;
    const int hi   = lane >> 4;
    const int m0   = blockIdx.x * 16;
    const int e0   = blockIdx.y * 64;
    const int n0   = wave * 16;
    const unsigned rows_rem = (unsigned)(PATCHES_TOTAL - m0);

    v8f c = {0.f, 0.f, 0.f, 0.f, 0.f, 0.f, 0.f, 0.f};

    if (wave == 0) {
        qt_tdm_load(qt_lds_off(&As[0][0]), patches + (size_t)m0 * K_DIM,
                    K_DIM, rows_rem, 0, 32, 16, 0, (unsigned long long)K_DIM,
                    0ull, true);
        qt_tdm_load(qt_lds_off(&Bs[0][0]), bw + (size_t)e0 * K_DIM,
                    K_DIM, (unsigned)(E_DIM - e0), 0, 32, 64, 0,
                    (unsigned long long)K_DIM, 0ull, true);
        __builtin_amdgcn_s_wait_tensorcnt(0);
    }
    __syncthreads();

    for (int k0 = 0; k0 < K_DIM; k0 += 32) {
        const int cur = (k0 >> 5) & 1;
        if (wave == 0 && k0 + 32 < K_DIM) {
            const int kn = k0 + 32;
            qt_tdm_load(qt_lds_off(&As[cur ^ 1][0]),
                        patches + (size_t)m0 * K_DIM + kn,
                        (unsigned)(K_DIM - kn), rows_rem, 0, 32, 16, 0,
                        (unsigned long long)K_DIM, 0ull, true);
            qt_tdm_load(qt_lds_off(&Bs[cur ^ 1][0]),
                        bw + (size_t)e0 * K_DIM + kn,
                        (unsigned)(K_DIM - kn), (unsigned)(E_DIM - e0), 0,
                        32, 64, 0, (unsigned long long)K_DIM, 0ull, true);
        }

        const float* Ac = As[cur];
        const float* Bc = Bs[cur];
        #pragma unroll
        for (int kk = 0; kk < 32; kk += 4) {
            v2f a, b;
            a.x = Ac[lm * 33 + kk + 2 * hi];
            a.y = Ac[lm * 33 + kk + 2 * hi + 1];
            b.x = Bc[(n0 + lm) * 33 + kk + 2 * hi];
            b.y = Bc[(n0 + lm) * 33 + kk + 2 * hi + 1];
            c = __builtin_amdgcn_wmma_f32_16x16x4_f32(
                    false, a, false, b, (short)0, c, false, false);
        }
        __syncthreads();                  // all waves done reading cur
        if (wave == 0) __builtin_amdgcn_s_wait_tensorcnt(0);
        __syncthreads();                  // next buffers visible
    }

    const int e = e0 + n0 + lm;
    #pragma unroll
    for (int r = 0; r < 8; ++r) {
        int row = m0 + r + 8 * hi;
        if (row < PATCHES_TOTAL)
            emb[(size_t)row * E_DIM + e] = c[r] + bb[e];
    }
}

// ---------------------------------------------------------------------------
// Kernel 4: per output cell, walk the tree: leaf = first level with logit<=0
// (or level 5). One 1024-thread block, one thread per 32x32 cell.
// ---------------------------------------------------------------------------
__global__ __launch_bounds__(1024) void qt_leaf(
    const float* __restrict__ logits, int* __restrict__ leaf)
{
    const int t  = threadIdx.x;
    const int gy = t >> 5, gx = t & 31;
    const int offs[6] = {0, 1, 5, 21, 85, 341};
    int res = 0;
    for (int l = 0; l <= 5; ++l) {
        int n = 1 << l;
        int i = gy >> (5 - l), j = gx >> (5 - l);
        int p = offs[l] + i * n + j;
        res = p;
        if (l == 5) break;
        if (!(logits[p] > 0.f)) break;   // NaN -> leaf, matches jnp semantics
    }
    leaf[t] = res;
}

// ---------------------------------------------------------------------------
// Kernel 5: scatter leaf embeddings to dense [512][32][32] output.
// ---------------------------------------------------------------------------
__global__ __launch_bounds__(256) void qt_gather_out(
    const float* __restrict__ emb, const int* __restrict__ leaf,
    float* __restrict__ out)
{
    int gid = blockIdx.x * 256 + threadIdx.x;
    if (gid >= E_DIM * 1024) return;
    int e    = gid >> 10;
    int cell = gid & 1023;
    out[gid] = emb[(size_t)leaf[cell] * E_DIM + e];
}

// ---------------------------------------------------------------------------
extern "C" void kernel_launch(void* const* d_in, const int* in_sizes, int n_in,
                              void* d_out, int out_size, void* d_ws, size_t ws_size,
                              hipStream_t stream)
{
    (void)in_sizes; (void)n_in; (void)out_size; (void)ws_size;

    const float* x   = (const float*)d_in[0];
    const float* c1w = (const float*)d_in[1];
    const float* c1b = (const float*)d_in[2];
    const float* g1  = (const float*)d_in[3];
    const float* b1  = (const float*)d_in[4];
    const float* m1  = (const float*)d_in[5];
    const float* v1  = (const float*)d_in[6];
    const float* c2w = (const float*)d_in[7];
    const float* c2b = (const float*)d_in[8];
    const float* g2  = (const float*)d_in[9];
    const float* b2  = (const float*)d_in[10];
    const float* m2  = (const float*)d_in[11];
    const float* v2  = (const float*)d_in[12];
    const float* c3w = (const float*)d_in[13];
    const float* c3b = (const float*)d_in[14];
    const float* g3  = (const float*)d_in[15];
    const float* b3  = (const float*)d_in[16];
    const float* m3  = (const float*)d_in[17];
    const float* v3  = (const float*)d_in[18];
    const float* lw  = (const float*)d_in[19];
    const float* lb  = (const float*)d_in[20];
    const float* bw  = (const float*)d_in[21];
    const float* bb  = (const float*)d_in[22];

    float* ws      = (float*)d_ws;
    float* patches = ws;                                        // 1365*12288
    float* emb     = patches + (size_t)PATCHES_TOTAL * K_DIM;   // 1365*512
    float* logits  = emb + (size_t)PATCHES_TOTAL * E_DIM;       // 1365 (pad 2048)
    int*   leaf    = (int*)(logits + 2048);                     // 1024

    qt_extract_patches<<<(PATCHES_TOTAL * K_DIM + 255) / 256, 256, 0, stream>>>(
        x, patches);

    // dynamic LDS: 18432 + 3072 + 5952 + 7680 + 1024 + 64 floats = 144,896 B
    const size_t splitter_lds = (size_t)(18432 + 3072 + 5952 + 7680 + 1024 + 64)
                                * sizeof(float);
    qt_splitter<<<PATCHES_TOTAL, 256, splitter_lds, stream>>>(
        patches, c1w, c1b, g1, b1, m1, v1, c2w, c2b, g2, b2, m2, v2,
        c3w, c3b, g3, b3, m3, v3, lw, lb, logits);

    qt_embed_gemm<<<dim3((PATCHES_TOTAL + 15) / 16, E_DIM / 64), 128, 0, stream>>>(
        patches, bw, bb, emb);

    qt_leaf<<<1, 1024, 0, stream>>>(logits, leaf);

    qt_gather_out<<<(E_DIM * 1024 + 255) / 256, 256, 0, stream>>>(
        emb, leaf, (float*)d_out);
}